// SLKA_893353198367
// MI455X (gfx1250) — compile-verified
//
#include <hip/hip_runtime.h>
#include <hip/hip_bf16.h>
#include <math.h>

// ---------------------------------------------------------------------------
// CDNA5 (gfx1250) WMMA helpers: dense fp32 path V_WMMA_F32_16X16X4_F32
// ---------------------------------------------------------------------------
typedef float v2f __attribute__((ext_vector_type(2)));
typedef float v4f __attribute__((ext_vector_type(4)));
typedef float v8f __attribute__((ext_vector_type(8)));

static __device__ __forceinline__ v8f wmma4(v2f a, v2f b, v8f c) {
  // 8 args: (neg_a, A, neg_b, B, c_mod, C, reuse_a, reuse_b)
  return __builtin_amdgcn_wmma_f32_16x16x4_f32(false, a, false, b,
                                               (short)0, c, false, false);
}

static __device__ __forceinline__ float gelu_f(float x) {
  return 0.5f * x * (1.0f + erff(x * 0.70710678118654752f));
}
static __device__ __forceinline__ float sigmoid_f(float x) {
  return 1.0f / (1.0f + __expf(-x));
}

// Problem constants
#define BQ 4
#define HH 64
#define WW 64
#define NPIX (BQ * HH * WW)   // 16384
#define FCH 96
#define CCH 192
#define KD 5
#define KKD (KD * KD)         // 25
#define PH 30                 // pooled spatial
#define MPOOL (BQ * PH * PH)  // 3600
#define LSTR 484              // padded im2col pixel stride (floats); 484%64=36
                              // -> 32 lanes tile all 64 LDS banks, b128-aligned

// ---------------------------------------------------------------------------
// Kernel 1: head 1x1 conv  (16384 x 192) * (192 x 96) -> h0p (p-major, 96)
// one wave = 16-row strip x all 6 N tiles
// ---------------------------------------------------------------------------
__global__ void k_head(const float* __restrict__ x, const float* __restrict__ hw,
                       const float* __restrict__ hb, float* __restrict__ h0p) {
  const int wave = threadIdx.x >> 5, lane = threadIdx.x & 31;
  const int half = lane >> 4, col = lane & 15;
  const int row0 = (blockIdx.x * 8 + wave) * 16;
  v8f acc[6];
#pragma unroll
  for (int t = 0; t < 6; ++t) acc[t] = (v8f)0.f;

  for (int k = 0; k < CCH; k += 4) {
    const float* ap = x + (size_t)(row0 + col) * CCH + k + 2 * half;
    v2f a; a.x = ap[0]; a.y = ap[1];
#pragma unroll
    for (int t = 0; t < 6; ++t) {
      const int n = t * 16 + col;
      const float* bp = hw + (size_t)n * CCH + k + 2 * half;
      v2f b; b.x = bp[0]; b.y = bp[1];
      acc[t] = wmma4(a, b, acc[t]);
    }
  }
#pragma unroll
  for (int t = 0; t < 6; ++t) {
    const int n = t * 16 + col;
    const float bias = hb[n];
#pragma unroll
    for (int i = 0; i < 8; ++i) {
      const int m = row0 + i + 8 * half;
      h0p[(size_t)m * FCH + n] = acc[t][i] + bias;
    }
  }
}

// ---------------------------------------------------------------------------
// Kernel 2: offset conv 5x5 pad2 (96 -> 50, padded to 64) implicit GEMM
// offp stride 64 per pixel, channels 50..63 are dead
// ---------------------------------------------------------------------------
__global__ void k_off(const float* __restrict__ h0p, const float* __restrict__ ow,
                      const float* __restrict__ ob, float* __restrict__ offp) {
  const int wave = threadIdx.x >> 5, lane = threadIdx.x & 31;
  const int half = lane >> 4, col = lane & 15;
  const int p0 = (blockIdx.x * 8 + wave) * 16;
  const int pm = p0 + col;
  const int h = (pm >> 6) & 63, w = pm & 63;
  v8f acc[4];
#pragma unroll
  for (int t = 0; t < 4; ++t) acc[t] = (v8f)0.f;

  for (int t = 0; t < KKD; ++t) {
    // warm next tap's weight line into GL2 (global_prefetch_b8)
    if (col < 50 / 16 + 1) {
      const int tn = (t + 1 < KKD) ? t + 1 : t;
      __builtin_prefetch(ow + ((size_t)(col * 16) * FCH) * KKD + tn, 0, 1);
    }
    const int ky = t / KD, kx = t % KD;
    const int hh = h + ky - 2, wp = w + kx - 2;
    const bool valid = (hh >= 0) & (hh < HH) & (wp >= 0) & (wp < WW);
    const float* abase = h0p + (size_t)(pm + (ky - 2) * WW + (kx - 2)) * FCH;
    for (int cb = 0; cb < FCH; cb += 4) {
      const int c = cb + 2 * half;
      v2f a;
      a.x = valid ? abase[c] : 0.f;
      a.y = valid ? abase[c + 1] : 0.f;
#pragma unroll
      for (int nt = 0; nt < 4; ++nt) {
        const int n = nt * 16 + col;
        v2f b;
        if (n < 50) {
          const float* bp = ow + ((size_t)n * FCH + c) * KKD + t;
          b.x = bp[0]; b.y = bp[KKD];
        } else { b.x = 0.f; b.y = 0.f; }
        acc[nt] = wmma4(a, b, acc[nt]);
      }
    }
  }
#pragma unroll
  for (int nt = 0; nt < 4; ++nt) {
    const int n = nt * 16 + col;
    const float bias = (n < 50) ? ob[n] : 0.f;
#pragma unroll
    for (int i = 0; i < 8; ++i) {
      const int m = p0 + i + 8 * half;
      offp[(size_t)m * 64 + n] = acc[nt][i] + bias;
    }
  }
}

// ---------------------------------------------------------------------------
// Kernel 3: deformable 5x5 conv. Block = 384 thr (12 waves), 32 pixels.
// Phase 1: bilinear-gather im2col (5-tap chunk) into LDS, b128 vectorized.
// Phase 2: WMMA from LDS (bank-conflict-free via LSTR=484 padding);
//          wave -> (m_sub in {0,1}, n_tile in {0..5}).
// ---------------------------------------------------------------------------
__global__ void k_deform(const float* __restrict__ h0p, const float* __restrict__ offp,
                         const float* __restrict__ dcw, const float* __restrict__ dcb,
                         float* __restrict__ c1p) {
  __shared__ float lds[32 * LSTR];      // 61,952 B
  const int tid = threadIdx.x;
  const int wave = tid >> 5, lane = tid & 31;
  const int half = lane >> 4, col = lane & 15;
  const int p0 = blockIdx.x * 32;
  const int n_tile = wave % 6, m_sub = wave / 6;
  v8f acc = (v8f)0.f;

  for (int cc = 0; cc < 5; ++cc) {
    __syncthreads();
    // -------- phase 1: build 32px x 5tap x 96 im2col tile --------
    for (int idx = tid; idx < 32 * 5; idx += blockDim.x) {
      const int pl = idx & 31, tl = idx >> 5;
      const int p = p0 + pl;
      const int tap = cc * 5 + tl;
      const int ky = tap / KD, kx = tap % KD;
      const int b = p >> 12, h = (p >> 6) & 63, w = p & 63;
      const float oy = offp[(size_t)p * 64 + tap * 2];
      const float ox = offp[(size_t)p * 64 + tap * 2 + 1];
      const float py = oy + (float)(h + ky - 2);
      const float px = ox + (float)(w + kx - 2);
      const float y0f = floorf(py), x0f = floorf(px);
      const float fy = py - y0f, fx = px - x0f;
      const int iy0 = (int)y0f, ix0 = (int)x0f;
      const int   ys[2] = { iy0, iy0 + 1 };
      const int   xs[2] = { ix0, ix0 + 1 };
      const float wy[2] = { 1.f - fy, fy };
      const float wx[2] = { 1.f - fx, fx };
      float cw[4]; const v4f* cbp[4];
      int q = 0;
#pragma unroll
      for (int ay = 0; ay < 2; ++ay)
#pragma unroll
        for (int ax = 0; ax < 2; ++ax) {
          const int yy = ys[ay], xx = xs[ax];
          const bool v = (yy >= 0) & (yy < HH) & (xx >= 0) & (xx < WW);
          const int yc = min(max(yy, 0), HH - 1), xc = min(max(xx, 0), WW - 1);
          cw[q] = v ? wy[ay] * wx[ax] : 0.f;
          cbp[q] = (const v4f*)(h0p + (size_t)(((b * HH) + yc) * WW + xc) * FCH);
          ++q;
        }
      v4f* dst = (v4f*)(lds + pl * LSTR + tl * FCH);
#pragma unroll 4
      for (int c4 = 0; c4 < FCH / 4; ++c4)
        dst[c4] = cw[0] * cbp[0][c4] + cw[1] * cbp[1][c4]
                + cw[2] * cbp[2][c4] + cw[3] * cbp[3][c4];
    }
    __syncthreads();
    // -------- phase 2: WMMA over this 480-deep K chunk --------
    const int mbase = m_sub * 16;
    const int n = n_tile * 16 + col;
    for (int kb = 0; kb < 5 * FCH; kb += 4) {
      const int tl = kb / FCH;
      const int c = (kb % FCH) + 2 * half;
      const float* ap = lds + (mbase + col) * LSTR + tl * FCH + c;
      v2f a; a.x = ap[0]; a.y = ap[1];
      const int tap = cc * 5 + tl;
      const float* bp = dcw + ((size_t)n * FCH + c) * KKD + tap;
      v2f b; b.x = bp[0]; b.y = bp[KKD];
      acc = wmma4(a, b, acc);
    }
  }
  const int n = n_tile * 16 + col;
  const float bias = dcb[n];
#pragma unroll
  for (int i = 0; i < 8; ++i) {
    const int m = p0 + m_sub * 16 + i + 8 * half;
    c1p[(size_t)m * FCH + n] = acc[i] + bias;
  }
}

// ---------------------------------------------------------------------------
// Kernel 4: maxpool 5x5 stride 2 VALID : (64x64) -> (30x30)
// ---------------------------------------------------------------------------
__global__ void k_pool(const float* __restrict__ c1p, float* __restrict__ c2) {
  const int idx = blockIdx.x * blockDim.x + threadIdx.x;
  if (idx >= MPOOL * FCH) return;
  const int c = idx % FCH;
  const int pix = idx / FCH;
  const int ow = pix % PH, oh = (pix / PH) % PH, b = pix / (PH * PH);
  float m = -INFINITY;
#pragma unroll
  for (int dy = 0; dy < 5; ++dy)
#pragma unroll
    for (int dx = 0; dx < 5; ++dx) {
      const int ih = oh * 2 + dy, iw = ow * 2 + dx;
      m = fmaxf(m, c1p[(size_t)(((b * HH) + ih) * WW + iw) * FCH + c]);
    }
  c2[(size_t)pix * FCH + c] = m;
}

// ---------------------------------------------------------------------------
// Kernel 5: generic 30x30 conv (96->96) implicit GEMM + activation
// act: 0=none 1=gelu 2=sigmoid
// ---------------------------------------------------------------------------
__global__ void k_conv30(const float* __restrict__ in, const float* __restrict__ wt,
                         const float* __restrict__ bias, float* __restrict__ out,
                         int ks, int pad, int act) {
  const int wave = threadIdx.x >> 5, lane = threadIdx.x & 31;
  const int half = lane >> 4, col = lane & 15;
  const int p0 = (blockIdx.x * 8 + wave) * 16;
  const int pm = p0 + col;
  const bool prow = pm < MPOOL;
  const int b = pm / (PH * PH), rem = pm % (PH * PH);
  const int h = rem / PH, w = rem % PH;
  const int KK = ks * ks;
  v8f acc[6];
#pragma unroll
  for (int t = 0; t < 6; ++t) acc[t] = (v8f)0.f;

  for (int t = 0; t < KK; ++t) {
    // warm next tap's weight stream (global_prefetch_b8)
    {
      const int tn = (t + 1 < KK) ? t + 1 : t;
      __builtin_prefetch(wt + ((size_t)(lane * 3) * FCH) * KK + tn, 0, 1);
    }
    const int ky = t / ks, kx = t % ks;
    const int hh = h + ky - pad, wp = w + kx - pad;
    const bool valid = prow & (hh >= 0) & (hh < PH) & (wp >= 0) & (wp < PH);
    const float* abase = in + (size_t)((b * PH * PH) + hh * PH + wp) * FCH;
    for (int cb = 0; cb < FCH; cb += 4) {
      const int c = cb + 2 * half;
      v2f a;
      a.x = valid ? abase[c] : 0.f;
      a.y = valid ? abase[c + 1] : 0.f;
#pragma unroll
      for (int nt = 0; nt < 6; ++nt) {
        const int n = nt * 16 + col;
        const float* bp = wt + ((size_t)n * FCH + c) * KK + t;
        v2f bb; bb.x = bp[0]; bb.y = bp[KK];
        acc[nt] = wmma4(a, bb, acc[nt]);
      }
    }
  }
#pragma unroll
  for (int nt = 0; nt < 6; ++nt) {
    const int n = nt * 16 + col;
    const float bv = bias[n];
#pragma unroll
    for (int i = 0; i < 8; ++i) {
      const int m = p0 + i + 8 * half;
      if (m < MPOOL) {
        float v = acc[nt][i] + bv;
        if (act == 1) v = gelu_f(v);
        else if (act == 2) v = sigmoid_f(v);
        out[(size_t)m * FCH + n] = v;
      }
    }
  }
}

// ---------------------------------------------------------------------------
// Kernel 6: bilinear resize 30x30 -> 64x64 (half-pixel, edge clamp)
// ---------------------------------------------------------------------------
__global__ void k_resize(const float* __restrict__ in, float* __restrict__ out) {
  const int idx = blockIdx.x * blockDim.x + threadIdx.x;
  if (idx >= NPIX * FCH) return;
  const int c = idx % FCH;
  const int p = idx / FCH;
  const int b = p >> 12, h = (p >> 6) & 63, w = p & 63;
  const float s = 30.f / 64.f;
  const float sy = (h + 0.5f) * s - 0.5f;
  const float sx = (w + 0.5f) * s - 0.5f;
  const float y0f = floorf(sy), x0f = floorf(sx);
  const float fy = sy - y0f, fx = sx - x0f;
  const int y0 = min(max((int)y0f, 0), PH - 1);
  const int y1 = min(max((int)y0f + 1, 0), PH - 1);
  const int x0 = min(max((int)x0f, 0), PH - 1);
  const int x1 = min(max((int)x0f + 1, 0), PH - 1);
  const size_t base = (size_t)b * PH * PH;
  const float v00 = in[(base + y0 * PH + x0) * FCH + c];
  const float v01 = in[(base + y0 * PH + x1) * FCH + c];
  const float v10 = in[(base + y1 * PH + x0) * FCH + c];
  const float v11 = in[(base + y1 * PH + x1) * FCH + c];
  out[(size_t)p * FCH + c] =
      (1.f - fy) * ((1.f - fx) * v00 + fx * v01) + fy * ((1.f - fx) * v10 + fx * v11);
}

// ---------------------------------------------------------------------------
// Kernel 7a: global mean of c1 over HxW per (b,c)
// ---------------------------------------------------------------------------
__global__ void k_gmean(const float* __restrict__ c1p, float* __restrict__ g0) {
  __shared__ float red[256];
  const int bid = blockIdx.x;        // = b*96 + c
  const int b = bid / FCH, c = bid % FCH;
  float s = 0.f;
  for (int i = threadIdx.x; i < HH * WW; i += blockDim.x)
    s += c1p[((size_t)b * HH * WW + i) * FCH + c];
  red[threadIdx.x] = s;
  __syncthreads();
  for (int st = 128; st > 0; st >>= 1) {
    if (threadIdx.x < st) red[threadIdx.x] += red[threadIdx.x + st];
    __syncthreads();
  }
  if (threadIdx.x == 0) g0[bid] = red[0] * (1.f / (HH * WW));
}

// ---------------------------------------------------------------------------
// Kernel 7b: g-branch MLP. 7x7/5x5 convs on 1x1 spatial collapse to center tap.
// ---------------------------------------------------------------------------
__global__ void k_gmlp(const float* __restrict__ g0,
                       const float* __restrict__ g1w, const float* __restrict__ g1b,
                       const float* __restrict__ g2w, const float* __restrict__ g2b,
                       const float* __restrict__ g3w, const float* __restrict__ g3b,
                       float* __restrict__ g3) {
  __shared__ float s0[FCH], s1[FCH], s2[FCH];
  const int b = blockIdx.x, t = threadIdx.x;
  if (t < FCH) s0[t] = g0[b * FCH + t];
  __syncthreads();
  if (t < FCH) {
    float s = g1b[t];
    for (int i = 0; i < FCH; ++i) s += s0[i] * g1w[((size_t)t * FCH + i) * 49 + 24];
    s1[t] = gelu_f(s);
  }
  __syncthreads();
  if (t < FCH) {
    float s = g2b[t];
    for (int i = 0; i < FCH; ++i) s += s1[i] * g2w[((size_t)t * FCH + i) * 25 + 12];
    s2[t] = gelu_f(s);
  }
  __syncthreads();
  if (t < FCH) {
    float s = g3b[t];
    for (int i = 0; i < FCH; ++i) s += s2[i] * g3w[((size_t)t * FCH + i) * 25 + 12];
    g3[b * FCH + t] = sigmoid_f(s);
  }
}

// ---------------------------------------------------------------------------
// Kernel 8: tail 1x1 (96 -> 192) on (c3+g), sigmoid, multiply by x, write out
// ---------------------------------------------------------------------------
__global__ void k_tail(const float* __restrict__ c3p, const float* __restrict__ g3,
                       const float* __restrict__ tw, const float* __restrict__ tb,
                       const float* __restrict__ x, float* __restrict__ out) {
  const int wave = threadIdx.x >> 5, lane = threadIdx.x & 31;
  const int half = lane >> 4, col = lane & 15;
  const int p0 = (blockIdx.x * 8 + wave) * 16;
  const int pm = p0 + col;
  const int bb = pm >> 12;
  v8f acc[12];
#pragma unroll
  for (int t = 0; t < 12; ++t) acc[t] = (v8f)0.f;

  for (int k = 0; k < FCH; k += 4) {
    const int kk = k + 2 * half;
    v2f a;
    a.x = c3p[(size_t)pm * FCH + kk]     + g3[bb * FCH + kk];
    a.y = c3p[(size_t)pm * FCH + kk + 1] + g3[bb * FCH + kk + 1];
#pragma unroll
    for (int nt = 0; nt < 12; ++nt) {
      const int n = nt * 16 + col;
      const float* bp = tw + (size_t)n * FCH + kk;
      v2f b; b.x = bp[0]; b.y = bp[1];
      acc[nt] = wmma4(a, b, acc[nt]);
    }
  }
#pragma unroll
  for (int nt = 0; nt < 12; ++nt) {
    const int n = nt * 16 + col;
    const float bias = tb[n];
#pragma unroll
    for (int i = 0; i < 8; ++i) {
      const int m = p0 + i + 8 * half;
      const float a = sigmoid_f(acc[nt][i] + bias);
      out[(size_t)m * CCH + n] = x[(size_t)m * CCH + n] * a;
    }
  }
}

// ---------------------------------------------------------------------------
// Launcher
// ---------------------------------------------------------------------------
extern "C" void kernel_launch(void* const* d_in, const int* in_sizes, int n_in,
                              void* d_out, int out_size, void* d_ws, size_t ws_size,
                              hipStream_t stream) {
  const float* x      = (const float*)d_in[0];
  const float* head_w = (const float*)d_in[1];
  const float* head_b = (const float*)d_in[2];
  const float* off_w  = (const float*)d_in[3];
  const float* off_b  = (const float*)d_in[4];
  const float* dc_w   = (const float*)d_in[5];
  const float* dc_b   = (const float*)d_in[6];
  const float* lka1_w = (const float*)d_in[7];
  const float* lka1_b = (const float*)d_in[8];
  const float* lka2_w = (const float*)d_in[9];
  const float* lka2_b = (const float*)d_in[10];
  const float* lka3_w = (const float*)d_in[11];
  const float* lka3_b = (const float*)d_in[12];
  const float* g1_w   = (const float*)d_in[13];
  const float* g1_b   = (const float*)d_in[14];
  const float* g2_w   = (const float*)d_in[15];
  const float* g2_b   = (const float*)d_in[16];
  const float* g3_w   = (const float*)d_in[17];
  const float* g3_b   = (const float*)d_in[18];
  const float* tail_w = (const float*)d_in[19];
  const float* tail_b = (const float*)d_in[20];
  float* out = (float*)d_out;

  float* ws = (float*)d_ws;
  float* H0P  = ws;                          // 16384*96
  float* OFFP = H0P  + (size_t)NPIX * FCH;   // 16384*64
  float* C1P  = OFFP + (size_t)NPIX * 64;    // 16384*96
  float* C2A  = C1P  + (size_t)NPIX * FCH;   // 3600*96
  float* C2B  = C2A  + (size_t)MPOOL * FCH;  // 3600*96
  float* C3P  = C2B  + (size_t)MPOOL * FCH;  // 16384*96
  float* G0   = C3P  + (size_t)NPIX * FCH;   // 4*96
  float* G3   = G0   + BQ * FCH;             // 4*96

  // 1. head 1x1 conv (WMMA GEMM)
  k_head<<<NPIX / 128, 256, 0, stream>>>(x, head_w, head_b, H0P);
  // 2. offset conv (WMMA implicit GEMM)
  k_off<<<NPIX / 128, 256, 0, stream>>>(H0P, off_w, off_b, OFFP);
  // 3. deformable conv (gather im2col -> LDS -> WMMA)
  k_deform<<<NPIX / 32, 384, 0, stream>>>(H0P, OFFP, dc_w, dc_b, C1P);
  // 4. maxpool 5x5 s2
  k_pool<<<(MPOOL * FCH + 255) / 256, 256, 0, stream>>>(C1P, C2A);
  // 5-7. LKA convs (WMMA) with fused activations
  const int cgrid = (MPOOL + 127) / 128;
  k_conv30<<<cgrid, 256, 0, stream>>>(C2A, lka1_w, lka1_b, C2B, 7, 3, 1);
  k_conv30<<<cgrid, 256, 0, stream>>>(C2B, lka2_w, lka2_b, C2A, 5, 2, 1);
  k_conv30<<<cgrid, 256, 0, stream>>>(C2A, lka3_w, lka3_b, C2B, 5, 2, 2);
  // 8. bilinear upsample 30 -> 64
  k_resize<<<(NPIX * FCH + 255) / 256, 256, 0, stream>>>(C2B, C3P);
  // 9. global-average + g MLP
  k_gmean<<<BQ * FCH, 256, 0, stream>>>(C1P, G0);
  k_gmlp<<<BQ, 128, 0, stream>>>(G0, g1_w, g1_b, g2_w, g2_b, g3_w, g3_b, G3);
  // 10. tail GEMM + sigmoid + elementwise (WMMA)
  k_tail<<<NPIX / 128, 256, 0, stream>>>(C3P, G3, tail_w, tail_b, x, out);
}